// TestAllReduceRMSNormModel_12592844112268
// MI455X (gfx1250) — compile-verified
//
#include <hip/hip_runtime.h>
#include <cstdint>

// Problem constants (from the reference): hs [TP=4, tokens, 4096] fp32.
#define HIDDEN  4096
#define EPS     1e-6f
#define BLOCK   256
#define WAVES   (BLOCK / 32)            // wave32 on gfx1250 -> 8 waves
#define CHUNKS  (HIDDEN / (4 * BLOCK))  // 4 float4 chunks per thread per slice

typedef float v4f __attribute__((ext_vector_type(4)));
typedef int   v4i __attribute__((vector_size(4 * sizeof(int))));  // matches builtin param

// gfx1250 async global->LDS copy path (ASYNCcnt-tracked), guarded so a missing
// builtin degrades to plain loads instead of breaking the compile.
#if defined(__has_builtin)
#  if __has_builtin(__builtin_amdgcn_global_load_async_to_lds_b128)
#    define HAVE_ASYNC_LDS 1
#  endif
#  if __has_builtin(__builtin_amdgcn_s_wait_asynccnt)
#    define HAVE_WAIT_ASYNC 1
#  endif
#endif

__global__ void __launch_bounds__(BLOCK) ar_rmsnorm_kernel(
    const float* __restrict__ hs,   // [4, tokens, HIDDEN] partial sums
    const float* __restrict__ w,    // [HIDDEN]
    float* __restrict__ out,        // [tokens, HIDDEN]
    long tokens)
{
    __shared__ float s_w[HIDDEN];   // 16 KB staged weight (reused by all threads)
    __shared__ float s_red[WAVES];

    const int  tid     = threadIdx.x;
    const long t       = blockIdx.x;
    const long slice   = tokens * (long)HIDDEN;   // elements per TP slice
    const long tok_off = t * (long)HIDDEN;

    // ---- Stage weight into LDS asynchronously; overlaps with the streaming
    // ---- accumulation loads below. Tracked by ASYNCcnt.
#if HAVE_ASYNC_LDS
    #pragma unroll
    for (int c = 0; c < CHUNKS; ++c) {
        const int idx = (c * BLOCK + tid) * 4;
        __builtin_amdgcn_global_load_async_to_lds_b128(
            (v4i*)(w + idx),
            (v4i*)(s_w + idx),
            /*offset=*/0, /*cpol=*/0);
    }
#else
    #pragma unroll
    for (int c = 0; c < CHUNKS; ++c) {
        const int idx = (c * BLOCK + tid) * 4;
        *(v4f*)(s_w + idx) = *(const v4f*)(w + idx);
    }
#endif

    // ---- All-reduce over TP=4 in registers; non-temporal (stream-once data,
    // ---- keep L2 for the weight). 16 x global_load_b128 per thread.
    v4f acc[CHUNKS];
    #pragma unroll
    for (int c = 0; c < CHUNKS; ++c) {
        const long base = tok_off + (long)((c * BLOCK + tid) * 4);
        v4f a0 = __builtin_nontemporal_load((const v4f*)(hs + base));
        v4f a1 = __builtin_nontemporal_load((const v4f*)(hs + slice     + base));
        v4f a2 = __builtin_nontemporal_load((const v4f*)(hs + 2 * slice + base));
        v4f a3 = __builtin_nontemporal_load((const v4f*)(hs + 3 * slice + base));
        acc[c] = (a0 + a1) + (a2 + a3);
    }

    // ---- Sum of squares (per thread), then wave32 butterfly reduce.
    float ss = 0.0f;
    #pragma unroll
    for (int c = 0; c < CHUNKS; ++c) {
        v4f q = acc[c] * acc[c];
        ss += (q.x + q.y) + (q.z + q.w);
    }
    #pragma unroll
    for (int m = 16; m >= 1; m >>= 1)
        ss += __shfl_xor(ss, m, 32);

    // ---- Make the async weight copies visible before the block barrier.
#if HAVE_ASYNC_LDS
#  if HAVE_WAIT_ASYNC
    __builtin_amdgcn_s_wait_asynccnt((short)0);
#  else
    asm volatile("s_wait_asynccnt 0" ::: "memory");
#  endif
#endif

    if ((tid & 31) == 0) s_red[tid >> 5] = ss;
    __syncthreads();

    float tot = 0.0f;
    #pragma unroll
    for (int i = 0; i < WAVES; ++i) tot += s_red[i];

    const float rstd = rsqrtf(tot * (1.0f / (float)HIDDEN) + EPS);

    // ---- Scale by rstd * weight (weight from LDS), non-temporal store.
    #pragma unroll
    for (int c = 0; c < CHUNKS; ++c) {
        const int idx = (c * BLOCK + tid) * 4;
        v4f wv = *(const v4f*)(s_w + idx);
        v4f o  = acc[c] * rstd * wv;
        __builtin_nontemporal_store(o, (v4f*)(out + tok_off + idx));
    }
}

extern "C" void kernel_launch(void* const* d_in, const int* in_sizes, int n_in,
                              void* d_out, int out_size, void* d_ws, size_t ws_size,
                              hipStream_t stream) {
    // setup_inputs order: hidden_states [4, tokens, 4096], residual [tokens, 4096]
    // (unused by the reference), weight [4096]. All fp32.
    const float* hs = (const float*)d_in[0];
    const float* w  = (const float*)d_in[2];
    float*       out = (float*)d_out;

    const long tokens = (long)in_sizes[1] / HIDDEN;  // residual is [tokens, HIDDEN]

    dim3 grid((unsigned)tokens);   // one workgroup per token row
    dim3 block(BLOCK);
    ar_rmsnorm_kernel<<<grid, block, 0, stream>>>(hs, w, out, tokens);
}